// FastSpeech2ConformerLengthRegulator_8486855377379
// MI455X (gfx1250) — compile-verified
//
#include <hip/hip_runtime.h>

// FastSpeech2Conformer length regulator for MI455X (gfx1250, wave32).
// Zero-FLOP, bandwidth-bound gather: ~215 MB HBM traffic -> ~9-10 us floor @ 23.3 TB/s.
// CDNA5-specific paths used:
//  - global_load_async_to_lds_b128 + s_wait_asynccnt: stage cum row memory->LDS
//    without a VGPR round trip (ASYNCcnt-tracked async copy engine)
//  - non-temporal global_store_b128: stream the write-once output past L2 so the
//    reusable 32 MB embedding table stays resident in the 192 MB L2
//  - global_prefetch_b8 lookahead on future phoneme rows
//  - wave32 shuffle scan (no barriers) for the duration cumsum

#define BB 32
#define TT 512
#define DD 512
#define FPB 64   // output frames per block tile

typedef __attribute__((ext_vector_type(4))) float v4f;
typedef __attribute__((ext_vector_type(4))) int   v4i;

// ---------------- Kernel A: gather / expand (hot kernel, ~215 MB moved) ----------------
// grid = (B, ceil(max_len/FPB)), block = 256 threads = 8 wave32s.
__global__ void lr_gather_kernel(const float* __restrict__ emb,
                                 const int* __restrict__ cum,
                                 const int* __restrict__ rowtotal,
                                 const int* __restrict__ flag,
                                 float* __restrict__ out,
                                 int max_len) {
    const int b = blockIdx.x;

    __shared__ int scum[TT];      // this row's cumulative offsets (2 KB)
    __shared__ int sidx[FPB];     // resolved phoneme index per frame (-1 = pad)

    const int wave = threadIdx.x >> 5;   // 0..7
    const int lane = threadIdx.x & 31;   // 0..31

    // --- async stage cum[b][0..511] into LDS (wave 0, 4 x 512B instructions) ---
    if (wave == 0) {
        // LDS byte offset = low 32 bits of the generic pointer (LDS aperture maps
        // the wave's LDS allocation at offset addr[31:0])
        unsigned lds0 = (unsigned)(size_t)(&scum[0]) + (unsigned)(lane * 16);
        unsigned long long g0 =
            (unsigned long long)(const char*)(cum + (size_t)b * TT) + (unsigned long long)(lane * 16);
        #pragma unroll
        for (int k = 0; k < 4; ++k) {
            unsigned l = lds0 + (unsigned)(k * 512);
            unsigned long long g = g0 + (unsigned long long)(k * 512);
            asm volatile("global_load_async_to_lds_b128 %0, %1, off"
                         :: "v"(l), "v"(g) : "memory");
        }
        asm volatile("s_wait_asynccnt 0x0" ::: "memory");
    }
    __syncthreads();

    const int allzero = flag[0];                    // uniform
    const int total   = allzero ? TT : rowtotal[b]; // row's valid frame count

    const int frame0 = blockIdx.y * FPB;
    const int fcount = min(FPB, max_len - frame0);  // frames in this tile

    // --- phase A: resolve phoneme index for every frame of the tile ---
    if (threadIdx.x < FPB) {
        const int f = frame0 + threadIdx.x;
        int idx = -1;                                // -1 -> pad with zeros
        if (threadIdx.x < fcount && f < total) {
            if (allzero) {
                idx = f;                             // all durations forced to 1
            } else {
                // branchless upper_bound over 512 sorted ints in LDS
                int lo = 0;
                #pragma unroll
                for (int s = 256; s > 0; s >>= 1) {
                    int m = lo + s - 1;
                    if (m < TT && scum[m] <= f) lo += s;
                }
                idx = min(lo, TT - 1);
            }
        }
        sidx[threadIdx.x] = idx;
    }
    __syncthreads();

    // --- phase B: tight copy stream, 2 frames per wave per iteration ---
    const size_t out_row = (size_t)b * (size_t)max_len;
    for (int j = wave; j < fcount; j += 16) {
        const int j1   = j + 8;
        const int has1 = (j1 < fcount);

        const int idx0 = sidx[j];
        const int idx1 = has1 ? sidx[j1] : -1;

        v4f a0 = {0.f, 0.f, 0.f, 0.f}, a1 = a0, a2 = a0, a3 = a0;
        v4f c0 = a0, c1 = a0, c2 = a0, c3 = a0;

        if (idx0 >= 0) {
            const v4f* src = (const v4f*)(emb + (((size_t)b * TT + idx0) * DD));
            a0 = src[lane];
            a1 = src[lane + 32];
            a2 = src[lane + 64];
            a3 = src[lane + 96];
        }
        if (idx1 >= 0) {
            const v4f* src = (const v4f*)(emb + (((size_t)b * TT + idx1) * DD));
            c0 = src[lane];
            c1 = src[lane + 32];
            c2 = src[lane + 64];
            c3 = src[lane + 96];
        }

        // prefetch the row we'll need next iteration (32 lanes x 64B = 2KB)
        const int jp = j + 16;
        if (jp < fcount) {
            const int idxp = sidx[jp];
            if (idxp >= 0) {
                const char* nxt =
                    (const char*)(emb + (((size_t)b * TT + idxp) * DD)) + lane * 64;
                __builtin_prefetch(nxt, 0, 3);
            }
        }

        // non-temporal streaming stores: output is write-once, never re-read
        v4f* dst0 = (v4f*)(out + (out_row + (size_t)(frame0 + j)) * DD);
        __builtin_nontemporal_store(a0, dst0 + lane);
        __builtin_nontemporal_store(a1, dst0 + lane + 32);
        __builtin_nontemporal_store(a2, dst0 + lane + 64);
        __builtin_nontemporal_store(a3, dst0 + lane + 96);
        if (has1) {
            v4f* dst1 = (v4f*)(out + (out_row + (size_t)(frame0 + j1)) * DD);
            __builtin_nontemporal_store(c0, dst1 + lane);
            __builtin_nontemporal_store(c1, dst1 + lane + 32);
            __builtin_nontemporal_store(c2, dst1 + lane + 64);
            __builtin_nontemporal_store(c3, dst1 + lane + 96);
        }
    }
}

// ---------------- Kernel B: fused duration cumsum + all-zero flag ----------------
// One block of (32,32): wave y scans batch row y with register+shuffle prefix
// (no barriers inside the scan); wave 0 then reduces row totals -> flag.
__global__ void lr_scan_flag_kernel(const int* __restrict__ dur,
                                    int* __restrict__ cum,
                                    int* __restrict__ rowtotal,
                                    int* __restrict__ flagp) {
    const int lane = threadIdx.x;   // 0..31
    const int b    = threadIdx.y;   // 0..31 (batch row, one wave per row)

    __shared__ int srow[BB];

    // each lane owns 16 contiguous durations: 4 x b128 loads
    const v4i* src = (const v4i*)(dur + (size_t)b * TT);
    int v[16];
    #pragma unroll
    for (int k = 0; k < 4; ++k)
        *(v4i*)(v + 4 * k) = src[lane * 4 + k];

    // in-register inclusive prefix over the 16 local values
    #pragma unroll
    for (int i = 1; i < 16; ++i) v[i] += v[i - 1];
    const int lsum = v[15];

    // wave32 inclusive scan of per-lane sums via shuffles (no LDS, no barriers)
    int x = lsum;
    #pragma unroll
    for (int off = 1; off < 32; off <<= 1) {
        int y = __shfl_up(x, off, 32);
        if (lane >= off) x += y;
    }
    const int excl = x - lsum;      // exclusive prefix for this lane's chunk

    #pragma unroll
    for (int i = 0; i < 16; ++i) v[i] += excl;

    v4i* dst = (v4i*)(cum + (size_t)b * TT);
    #pragma unroll
    for (int k = 0; k < 4; ++k)
        dst[lane * 4 + k] = *(v4i*)(v + 4 * k);

    if (lane == 31) {               // x == row inclusive total
        rowtotal[b] = x;
        srow[b]     = x;
    }
    __syncthreads();

    // wave 0: reduce the 32 row totals -> global all-zero flag
    if (b == 0) {
        int t = srow[lane];
        #pragma unroll
        for (int off = 16; off > 0; off >>= 1)
            t += __shfl_down(t, off, 32);
        if (lane == 0) *flagp = (t == 0) ? 1 : 0;
    }
}

// ---------------- Host launcher ----------------
extern "C" void kernel_launch(void* const* d_in, const int* in_sizes, int n_in,
                              void* d_out, int out_size, void* d_ws, size_t ws_size,
                              hipStream_t stream) {
    const float* emb = (const float*)d_in[0];   // [B, T, D] f32
    const int*   dur = (const int*)d_in[1];     // [B, T] i32
    float*       out = (float*)d_out;           // [B, max_len, D] f32

    // workspace layout: cum[B*T] | rowtotal[B] | flag[1]
    int* cum      = (int*)d_ws;
    int* rowtotal = cum + BB * TT;
    int* flag     = rowtotal + BB;

    const int max_len = out_size / (BB * DD);

    lr_scan_flag_kernel<<<dim3(1), dim3(32, 32), 0, stream>>>(dur, cum, rowtotal, flag);

    dim3 grid(BB, (max_len + FPB - 1) / FPB);
    lr_gather_kernel<<<grid, dim3(256), 0, stream>>>(emb, cum, rowtotal, flag, out, max_len);
}